// DeformConv3d_49100066128568
// MI455X (gfx1250) — compile-verified
//
#include <hip/hip_runtime.h>
#include <math.h>

// ---------------- problem constants (match reference) ----------------
#define Dd    16
#define Hh    128
#define Ww    128
#define NCIN  16
#define NCOUT 32
#define KTAPS 27          // 3*3*3
#define PADD  18          // Dd + 2
#define PADH  130
#define PADW  130
#define NVOX  (Dd*Hh*Ww)  // 262144

typedef __attribute__((ext_vector_type(2))) float v2f;
typedef __attribute__((ext_vector_type(8))) float v8f;
typedef __attribute__((ext_vector_type(4))) int   v4i;

#if defined(__has_builtin)
#if __has_builtin(__builtin_amdgcn_global_load_async_to_lds_b128) && \
    __has_builtin(__builtin_amdgcn_s_wait_asynccnt)
#define HAVE_ASYNC_LDS 1
#endif
#endif

#ifdef HAVE_ASYNC_LDS
typedef __attribute__((address_space(1))) v4i gv4i_t;  // global int4
typedef __attribute__((address_space(3))) v4i lv4i_t;  // LDS int4
#endif

// ---------------------------------------------------------------------
// Kernel 0: pad (+1 ring of zeros) and transpose x from [C][D][H][W]
// to channel-innermost [zp][yp][xp][C] so trilinear corner gathers are
// contiguous 64-byte blocks. 19.5 MB, lives in d_ws (L2-resident).
// ---------------------------------------------------------------------
__global__ void pad_transpose_kernel(const float* __restrict__ x,
                                     float* __restrict__ xp) {
  int idx = blockIdx.x * blockDim.x + threadIdx.x;
  const int total = PADD * PADH * PADW * NCIN;
  if (idx >= total) return;
  int c = idx & (NCIN - 1);
  int t = idx >> 4;
  int xpos = t % PADW; t /= PADW;
  int ypos = t % PADH; t /= PADH;
  int zpos = t;
  int xi = xpos - 1, yi = ypos - 1, zi = zpos - 1;
  float v = 0.0f;
  if (xi >= 0 && xi < Ww && yi >= 0 && yi < Hh && zi >= 0 && zi < Dd)
    v = x[((c * Dd + zi) * Hh + yi) * Ww + xi];
  xp[idx] = v;
}

// ---------------------------------------------------------------------
// Kernel 1: fused  (offset conv -> tanh offsets -> trilinear sample ->
// output contraction).  One wave32 per 16-voxel tile (tiles never cross
// a W row since 16 | 128).  Both GEMMs use V_WMMA_F32_16X16X4_F32.
// Phase-1 activations are staged into LDS via async global->LDS copies.
// ---------------------------------------------------------------------
__global__ void __launch_bounds__(32)
deform_fused_kernel(const float* __restrict__ xp,     // [18][130][130][16]
                    const float* __restrict__ weight, // [32][16][27]
                    const float* __restrict__ bias,   // [32]
                    const float* __restrict__ offw,   // [81][16][27]
                    const float* __restrict__ offb,   // [81]
                    float* __restrict__ out)          // [32][16][128][128]
{
  __shared__ float xs[3][3][18][NCIN];  // 3x3x18x16 slab for phase 1 (10.4 KB)
  __shared__ float soff[16][81];        // per-voxel tanh-scaled offsets

  const int lane = threadIdx.x;
  const int n    = lane & 15;      // N index (voxel in tile) / M row for A
  const int hi   = lane >> 4;      // lane half: selects K pair + C-row group
  const int hi2  = hi << 1;

  const int vb = blockIdx.x * 16;          // first voxel of tile
  const int z  = vb / (Hh * Ww);
  const int y  = (vb / Ww) % Hh;
  const int x0 = vb % Ww;
  const int xn = x0 + n;                   // this lane's voxel x

  // ---- Stage the 3x3x18x16 input slab into LDS (async copy path) ----
  // 9 rows of 18*16 floats; flat i: row = i/72, 16B chunk = i%72.
  {
    const size_t rb0 =
        (((size_t)z * PADH + y) * PADW + x0) << 4;  // floats, (kd=0,kh=0)
    for (int i = lane; i < 648; i += 32) {
      const int row = i / 72;               // 0..8 = kd*3+kh
      const int within = i - row * 72;      // 16-byte chunk in row
      const int kd = row / 3, kh = row - kd * 3;
      const float* g =
          xp + rb0 + ((size_t)kd * PADH + kh) * (PADW * NCIN) + within * 4;
      float* l = &xs[0][0][0][0] + i * 4;
#ifdef HAVE_ASYNC_LDS
      __builtin_amdgcn_global_load_async_to_lds_b128(
          (gv4i_t*)g, (lv4i_t*)l, 0, 0);
#else
      *(float4*)l = *(const float4*)g;
#endif
    }
#ifdef HAVE_ASYNC_LDS
    __builtin_amdgcn_s_wait_asynccnt(0);
#endif
    __syncthreads();
  }

  // ================= Phase 1: offset conv (96x16x432 via WMMA) ========
  // K order: k = tap*16 + cin.  A = offw (lane = M row), B = slab (lane = N).
  v8f acc[6];
#pragma unroll
  for (int i = 0; i < 6; ++i) acc[i] = (v8f){0,0,0,0,0,0,0,0};

  const int aoff_lane = n * 432 + hi2 * 27;   // float index into offw

  for (int t = 0; t < KTAPS; ++t) {
    const int kd = t / 9, kh = (t / 3) % 3, kw = t % 3;
    const float* bp = &xs[kd][kh][n + kw][hi2];
#pragma unroll
    for (int jc = 0; jc < 4; ++jc) {
      v2f b = *(const v2f*)(bp + jc * 4);
      const int ak = aoff_lane + jc * 108 + t;
#pragma unroll
      for (int mt = 0; mt < 5; ++mt) {      // m = mt*16+n <= 79 always valid
        const float* ap = offw + ak + mt * 6912;
        v2f a; a.x = ap[0]; a.y = ap[KTAPS];
        acc[mt] = __builtin_amdgcn_wmma_f32_16x16x4_f32(
            false, a, false, b, (short)0, acc[mt], false, false);
      }
      {                                      // tile 5: only m=80 (lane n==0)
        v2f a = (v2f){0.0f, 0.0f};
        if (n == 0) {
          const float* ap = offw + ak + 5 * 6912;
          a.x = ap[0]; a.y = ap[KTAPS];
        }
        acc[5] = __builtin_amdgcn_wmma_f32_16x16x4_f32(
            false, a, false, b, (short)0, acc[5], false, false);
      }
    }
  }

  // Epilogue: bias + tanh*scale, park offsets in LDS.
  // C/D layout: element r -> M = mt*16 + r + hi*8, N = n.
#pragma unroll
  for (int mt = 0; mt < 6; ++mt) {
#pragma unroll
    for (int r = 0; r < 8; ++r) {
      const int m = mt * 16 + r + hi * 8;
      if (m < 81) {
        const int comp = m % 3;                        // 0:dx 1:dy 2:dz
        const float scale = (comp == 2) ? 2.0f : 4.0f; // MAX_T / MAX_XY
        soff[n][m] = tanhf(acc[mt][r] + offb[m]) * scale;
      }
    }
  }
  __syncthreads();

  // ====== Phase 2: trilinear sample + output GEMM (32x16x432) =========
  v8f oacc[2];
  oacc[0] = (v8f){0,0,0,0,0,0,0,0};
  oacc[1] = (v8f){0,0,0,0,0,0,0,0};

  for (int t = 0; t < KTAPS; ++t) {
    // geometry for (voxel n, tap t): computed once, reused for 4 chunks
    const float dx = soff[n][t * 3 + 0];
    const float dy = soff[n][t * 3 + 1];
    const float dz = soff[n][t * 3 + 2];
    const int rx = t % 3 - 1, ry = (t / 3) % 3 - 1, rz = t / 9 - 1;
    const float gx = (float)(xn + 1 + rx) + dx;
    const float gy = (float)(y  + 1 + ry) + dy;
    const float gz = (float)(z  + 1 + rz) + dz;
    const float fx = floorf(gx), fy = floorf(gy), fz = floorf(gz);
    const float wx = gx - fx, wy = gy - fy, wz = gz - fz;
    const int ix = (int)fx, iy = (int)fy, iz = (int)fz;

    int   coff[8];
    float cw[8];
#pragma unroll
    for (int cr = 0; cr < 8; ++cr) {
      const int ddx = cr & 1, ddy = (cr >> 1) & 1, ddz = (cr >> 2) & 1;
      const int cx = ix + ddx, cy = iy + ddy, cz = iz + ddz;
      const bool valid = (cx >= 0) & (cx < PADW) & (cy >= 0) & (cy < PADH) &
                         (cz >= 0) & (cz < PADD);
      const int cxc = min(max(cx, 0), PADW - 1);
      const int cyc = min(max(cy, 0), PADH - 1);
      const int czc = min(max(cz, 0), PADD - 1);
      coff[cr] = ((czc * PADH + cyc) * PADW + cxc) << 4;
      const float w = (ddx ? wx : 1.0f - wx) *
                      (ddy ? wy : 1.0f - wy) *
                      (ddz ? wz : 1.0f - wz);
      cw[cr] = valid ? w : 0.0f;     // clip + mask, matches reference
    }

#pragma unroll
    for (int jc = 0; jc < 4; ++jc) { // 4 K-chunks per tap (16 channels)
      const int cb = jc * 4 + hi2;
      float s0 = 0.0f, s1 = 0.0f;
#pragma unroll
      for (int cr = 0; cr < 8; ++cr) {
        const float* p = xp + coff[cr] + cb;   // contiguous channel pair
        const float w = cw[cr];
        s0 = fmaf(w, p[0], s0);
        s1 = fmaf(w, p[1], s1);
      }
      v2f b = {s0, s1};
      const int ak = aoff_lane + jc * 108 + t;
#pragma unroll
      for (int mt = 0; mt < 2; ++mt) {
        const float* ap = weight + ak + mt * 6912;  // o = mt*16+n
        v2f a; a.x = ap[0]; a.y = ap[KTAPS];
        oacc[mt] = __builtin_amdgcn_wmma_f32_16x16x4_f32(
            false, a, false, b, (short)0, oacc[mt], false, false);
      }
    }
  }

  // Epilogue: D layout element r -> o = mt*16 + r + hi*8, voxel = vb + n.
#pragma unroll
  for (int mt = 0; mt < 2; ++mt) {
#pragma unroll
    for (int r = 0; r < 8; ++r) {
      const int o = mt * 16 + r + hi * 8;
      out[(size_t)o * NVOX + (vb + n)] = oacc[mt][r] + bias[o];
    }
  }
}

// ---------------------------------------------------------------------
extern "C" void kernel_launch(void* const* d_in, const int* in_sizes, int n_in,
                              void* d_out, int out_size, void* d_ws, size_t ws_size,
                              hipStream_t stream) {
  const float* x      = (const float*)d_in[0];  // [1,16,16,128,128]
  const float* weight = (const float*)d_in[1];  // [32,16,27]
  const float* bias   = (const float*)d_in[2];  // [32]
  const float* offw   = (const float*)d_in[3];  // [81,16,3,3,3]
  const float* offb   = (const float*)d_in[4];  // [81]
  float* out = (float*)d_out;                   // [1,32,16,128,128]
  float* xp  = (float*)d_ws;                    // padded/transposed x, 19.5 MB

  {
    const int total = PADD * PADH * PADW * NCIN;
    pad_transpose_kernel<<<(total + 255) / 256, 256, 0, stream>>>(x, xp);
  }
  deform_fused_kernel<<<NVOX / 16, 32, 0, stream>>>(xp, weight, bias, offw,
                                                    offb, out);
}